// Fold_45174466019401
// MI455X (gfx1250) — compile-verified
//
#include <hip/hip_runtime.h>

// Fold (col2im): x (8,32,8,8,4096) f32 -> out (8,32,260,260) f32
// kernel 8x8, stride 4x4, patch grid 64x64.
// One block per (b, c, output-row p). Stage the <=16 contiguous 64-float
// input segments for that row in LDS via async global->LDS copies, then
// each output element sums its <=4 contributors from LDS.

#define FB   8
#define FC   32
#define FK   8
#define FN   64
#define FH   260
// input strides in floats
#define STR_DJ 4096          // k2 stride (n*n)
#define STR_DI 32768         // k1 stride (8*4096)
#define STR_BC 262144        // per-(b,c) block (8*8*4096)

#if defined(__HIP_DEVICE_COMPILE__)
#if __has_builtin(__builtin_amdgcn_global_load_async_to_lds_b128)
#define FOLD_HAVE_ASYNC 1
#endif
#endif

typedef int v4i __attribute__((vector_size(16)));
typedef __attribute__((address_space(1))) v4i glb_v4i;
typedef __attribute__((address_space(3))) v4i lds_v4i;

__global__ __launch_bounds__(256) void fold_row_kernel(
    const float* __restrict__ x, float* __restrict__ out) {
  // lds[ii][dj][j] : ii indexes the (at most 2) valid i-values for this row
  __shared__ float lds[2][8][64];

  const int p   = blockIdx.x;   // output row, 0..259
  const int c   = blockIdx.y;
  const int b   = blockIdx.z;
  const int tid = threadIdx.x;

  // valid i range for row p:  p-7 <= 4i <= p  ->  i in [ceil((p-7)/4), floor(p/4)]
  int i_lo = (p - 4) >> 2;            // == floor((p-4)/4) == ceil((p-7)/4)
  if (i_lo < 0) i_lo = 0;
  int i_hi = p >> 2;
  if (i_hi > FN - 1) i_hi = FN - 1;
  const int ni = i_hi - i_lo + 1;     // 1 or 2

  const float* xbc = x + (size_t)(b * FC + c) * (size_t)STR_BC;

  // ---- stage: ni*8 segments of 64 contiguous floats, one float4 per lane ----
  const int nvec = ni * 8 * 16;       // 128 or 256 float4 transfers
  if (tid < nvec) {
    const int ii = tid >> 7;          // which i
    const int dj = (tid >> 4) & 7;
    const int j4 = tid & 15;          // float4 index within the 64-float row
    const int i  = i_lo + ii;
    const int di = p - 4 * i;
    const float* gptr =
        xbc + (size_t)di * STR_DI + (size_t)dj * STR_DJ + i * FN + 4 * j4;
    float* lptr = &lds[ii][dj][4 * j4];
#if defined(FOLD_HAVE_ASYNC)
    __builtin_amdgcn_global_load_async_to_lds_b128(
        (glb_v4i*)gptr, (lds_v4i*)lptr, /*imm offset*/ 0, /*cpol*/ 0);
#else
    *(float4*)lptr = *(const float4*)gptr;
#endif
  }
#if defined(FOLD_HAVE_ASYNC)
#if __has_builtin(__builtin_amdgcn_s_wait_asynccnt)
  __builtin_amdgcn_s_wait_asynccnt(0);
#else
  asm volatile("s_wait_asynccnt 0" ::: "memory");
#endif
#endif
  __syncthreads();

  // ---- compute: q = 4t + u.  Contributors (uniform across lanes):
  //   j = t   -> dj = u     (valid when t <= 63)
  //   j = t-1 -> dj = u + 4 (valid when t >= 1)
  if (tid < 65) {
    const int t = tid;
    float r[4];
#pragma unroll
    for (int u = 0; u < 4; ++u) {
      float s = 0.0f;
      if (t <= FN - 1) {
        s += lds[0][u][t];
        if (ni == 2) s += lds[1][u][t];
      }
      if (t >= 1) {
        s += lds[0][u + 4][t - 1];
        if (ni == 2) s += lds[1][u + 4][t - 1];
      }
      r[u] = s;
    }
    float* orow = out + ((size_t)(b * FC + c) * FH + p) * (size_t)FH;
    float4 v;
    v.x = r[0]; v.y = r[1]; v.z = r[2]; v.w = r[3];
    *(float4*)(orow + 4 * t) = v;   // row stride 1040B is 16B-aligned
  }
}

extern "C" void kernel_launch(void* const* d_in, const int* in_sizes, int n_in,
                              void* d_out, int out_size, void* d_ws, size_t ws_size,
                              hipStream_t stream) {
  (void)in_sizes; (void)n_in; (void)out_size; (void)d_ws; (void)ws_size;
  const float* x = (const float*)d_in[0];
  float* out = (float*)d_out;
  dim3 grid(FH, FC, FB);   // (row p, c, b) = (260, 32, 8)
  dim3 block(256);
  fold_row_kernel<<<grid, block, 0, stream>>>(x, out);
}